// SpatialEncoder_1649267441944
// MI455X (gfx1250) — compile-verified
//
#include <hip/hip_runtime.h>
#include <hip/hip_bf16.h>

// SpatialEncoder for MI455X (gfx1250): fp32 end-to-end using V_WMMA_F32_16X16X4_F32.
// v2: N-register-blocked WMMA tiles (NB accumulators per wave, A fragment reused
// NB times) + weights repacked into WMMA-chunk layout [k/4][half][OC][2] so every
// B fragment is one coalesced aligned b64 load.

typedef __attribute__((ext_vector_type(2))) float v2f;
typedef __attribute__((ext_vector_type(8))) float v8f;

#define WAVE 32

__device__ __forceinline__ v8f wmma4(v2f a, v2f b, v8f c) {
  // D = A(16x4) * B(4x16) + C(16x16), all fp32
  return __builtin_amdgcn_wmma_f32_16x16x4_f32(false, a, false, b, (short)0, c,
                                               false, false);
}

// ---------------------------------------------------------------------------
// Weight repack: OIHW -> packed [chunk][half][OC][2] with k = (kh*KW+kw)*IC+ic.
// Zero-pads k >= K (Kpad = ceil(K/4)*4).
// ---------------------------------------------------------------------------
__global__ void repack_w_k(const float* __restrict__ w, float* __restrict__ wp,
                           int IC, int OC, int KH, int KW, int Kpad) {
  int i = blockIdx.x * blockDim.x + threadIdx.x;
  if (i >= Kpad * OC) return;
  int oc = i % OC, k = i / OC;
  int K = KH * KW * IC;
  float v = 0.f;
  if (k < K) {
    int t  = k / IC, ic = k - t * IC;
    int kh = t / KW, kw = t - kh * KW;
    v = w[(((size_t)oc * IC + ic) * KH + kh) * KW + kw];
  }
  wp[(((size_t)(k >> 2) * 2 + ((k >> 1) & 1)) * OC + oc) * 2 + (k & 1)] = v;
}

// ---------------------------------------------------------------------------
// 1) Entity GEMM: reduced[32768,32] = relu(E[32768,256] x W1^T), NB=2
// ---------------------------------------------------------------------------
__global__ void entity_gemm_k(const float* __restrict__ E,
                              const float* __restrict__ wp,   // packed, OC=32
                              float* __restrict__ red) {
  const int Mtiles = 2048;
  int wid  = (blockIdx.x * blockDim.x + threadIdx.x) / WAVE;
  int lane = threadIdx.x & (WAVE - 1);
  if (wid >= Mtiles) return;
  int row0 = wid << 4;
  int half = lane >> 4, n = lane & 15;

  const float* Arow = E + (size_t)(row0 + n) * 256;
  v8f acc0 = {}, acc1 = {};
  for (int chunk = 0; chunk < 64; ++chunk) {
    int k = (chunk << 2) + 2 * half;
    v2f a = *(const v2f*)(Arow + k);
    const float* bp = wp + (size_t)(chunk * 2 + half) * 64;  // OC*2 = 64
    v2f b0 = *(const v2f*)(bp + n * 2);
    v2f b1 = *(const v2f*)(bp + (16 + n) * 2);
    acc0 = wmma4(a, b0, acc0);
    acc1 = wmma4(a, b1, acc1);
  }
#pragma unroll
  for (int i = 0; i < 8; ++i) {
    int m = row0 + i + 8 * half;
    float v0 = acc0[i]; v0 = v0 > 0.f ? v0 : 0.f;
    float v1 = acc1[i]; v1 = v1 > 0.f ? v1 : 0.f;
    red[(size_t)m * 32 + n]      = v0;
    red[(size_t)m * 32 + 16 + n] = v1;
  }
}

// ---------------------------------------------------------------------------
// 2) Scatter-add entities into map (NHWC [B,64,64,32]); scat pre-zeroed
// ---------------------------------------------------------------------------
__global__ void scatter_k(const int* __restrict__ xy,
                          const float* __restrict__ red,
                          float* __restrict__ scat) {
  int ent = blockIdx.x;       // 0 .. B*512-1
  int c   = threadIdx.x;      // 0 .. 31
  const int* bits = xy + (size_t)ent * 16;
  if (bits[0] == -1000000000) return;   // invalid entity
  int xv = 0, yv = 0;
#pragma unroll
  for (int j = 0; j < 8; ++j) {
    xv += bits[j]     << (7 - j);
    yv += bits[8 + j] << (7 - j);
  }
  xv /= 4; yv /= 4;                     // 0..63
  int b = ent >> 9;                     // /512
  int pix = yv * 64 + xv;
  atomicAdd(&scat[((size_t)(b << 12) + pix) * 32 + c],
            red[(size_t)ent * 32 + c]);
}

// ---------------------------------------------------------------------------
// 3) Projection 1x1 conv: concat(scat NHWC 32ch, x NCHW 18ch) -> 32ch, ReLU
//    M = 64*4096, N = 32, K = 50 (Kpad = 52, 13 chunks), NB=2
// ---------------------------------------------------------------------------
__device__ __forceinline__ float proj_ldA(const float* scat, const float* x,
                                          int mA, int bA, int pixA, int k) {
  if (k < 32)      return scat[(size_t)mA * 32 + k];
  else if (k < 50) return x[(((size_t)bA * 18 + (k - 32)) << 12) + pixA];
  else             return 0.f;
}

__global__ void project_k(const float* __restrict__ scat,
                          const float* __restrict__ x,
                          const float* __restrict__ wp,  // packed, OC=32, Kpad=52
                          const float* __restrict__ bias,
                          float* __restrict__ out) {     // NHWC [B,64,64,32]
  const int Mtiles = 262144 / 16;
  int wid  = (blockIdx.x * blockDim.x + threadIdx.x) / WAVE;
  int lane = threadIdx.x & 31;
  if (wid >= Mtiles) return;
  int row0 = wid << 4;
  int half = lane >> 4, n = lane & 15;
  int mA = row0 + n;
  int bA = mA >> 12, pixA = mA & 4095;

  v8f acc0 = {}, acc1 = {};
  for (int chunk = 0; chunk < 13; ++chunk) {
    int k = (chunk << 2) + 2 * half;
    v2f a;
    a.x = proj_ldA(scat, x, mA, bA, pixA, k);
    a.y = proj_ldA(scat, x, mA, bA, pixA, k + 1);
    const float* bp = wp + (size_t)(chunk * 2 + half) * 64;  // OC*2 = 64
    v2f b0 = *(const v2f*)(bp + n * 2);
    v2f b1 = *(const v2f*)(bp + (16 + n) * 2);
    acc0 = wmma4(a, b0, acc0);
    acc1 = wmma4(a, b1, acc1);
  }
#pragma unroll
  for (int i = 0; i < 8; ++i) {
    int m = row0 + i + 8 * half;
    float v0 = acc0[i] + bias[n];      v0 = v0 > 0.f ? v0 : 0.f;
    float v1 = acc1[i] + bias[16 + n]; v1 = v1 > 0.f ? v1 : 0.f;
    out[(size_t)m * 32 + n]      = v0;
    out[(size_t)m * 32 + 16 + n] = v1;
  }
}

// ---------------------------------------------------------------------------
// 4) Generic implicit-GEMM conv, NHWC activations, packed weights, NB tiles/wave.
//    IC % 4 == 0; OC % (16*NB) == 0; M = B*OH*OW multiple of 16.
// ---------------------------------------------------------------------------
template <int NB>
__global__ void conv_k(const float* __restrict__ in, int IH, int IW, int IC,
                       const float* __restrict__ wp, int OC,
                       const float* __restrict__ bias,
                       float* __restrict__ out, int OH, int OW,
                       int KH, int KW, int stride, int pad, int doRelu,
                       int Mtot) {
  int Ngroups = OC / (16 * NB);
  int wid  = (blockIdx.x * blockDim.x + threadIdx.x) / WAVE;
  int lane = threadIdx.x & 31;
  int totWaves = (Mtot >> 4) * Ngroups;
  if (wid >= totWaves) return;
  int tm = wid / Ngroups, tg = wid - tm * Ngroups;
  int row0 = tm << 4, col0 = tg * 16 * NB;
  int half = lane >> 4, n = lane & 15;

  int mA  = row0 + n;
  int b   = mA / (OH * OW);
  int pix = mA - b * (OH * OW);
  int oh  = pix / OW, ow = pix - oh * OW;
  int ICc = IC >> 2;                       // chunks per filter tap

  v8f acc[NB];
#pragma unroll
  for (int j = 0; j < NB; ++j) acc[j] = (v8f){};

  for (int kh = 0; kh < KH; ++kh) {
    int ih = oh * stride - pad + kh;
    bool okh = (ih >= 0) && (ih < IH);
    for (int kw = 0; kw < KW; ++kw) {
      int iw = ow * stride - pad + kw;
      bool ok = okh && (iw >= 0) && (iw < IW);
      const float* ap = in + (((size_t)b * IH + ih) * IW + iw) * IC;
      int base_chunk = (kh * KW + kw) * ICc;
      for (int icc = 0; icc < ICc; ++icc) {
        int ic = (icc << 2) + 2 * half;
        v2f av;
        if (ok) av = *(const v2f*)(ap + ic);
        else    { av.x = 0.f; av.y = 0.f; }
        const float* bp = wp + ((size_t)(base_chunk + icc) * 2 + half) * OC * 2;
#pragma unroll
        for (int j = 0; j < NB; ++j) {
          v2f bv = *(const v2f*)(bp + (col0 + j * 16 + n) * 2);
          acc[j] = wmma4(av, bv, acc[j]);
        }
      }
    }
  }
#pragma unroll
  for (int j = 0; j < NB; ++j) {
    int oc = col0 + j * 16 + n;
    float badd = bias ? bias[oc] : 0.f;
#pragma unroll
    for (int i = 0; i < 8; ++i) {
      int m = row0 + i + 8 * half;
      float v = acc[j][i] + badd;
      if (doRelu) v = v > 0.f ? v : 0.f;
      out[(size_t)m * OC + oc] = v;
    }
  }
}

// ---------------------------------------------------------------------------
// 5) BatchNorm (training-mode batch stats) over rows of t[M, C]
// ---------------------------------------------------------------------------
__global__ void bn_stats_k(const float* __restrict__ t, float* __restrict__ mean,
                           float* __restrict__ rstd, int Mrows, int C) {
  int c = blockIdx.x;
  float s = 0.f, s2 = 0.f;
  for (int m = threadIdx.x; m < Mrows; m += blockDim.x) {
    float v = t[(size_t)m * C + c];
    s += v; s2 += v * v;
  }
  __shared__ float sh[256], sh2[256];
  sh[threadIdx.x] = s; sh2[threadIdx.x] = s2;
  __syncthreads();
  for (int o = blockDim.x >> 1; o > 0; o >>= 1) {
    if ((int)threadIdx.x < o) {
      sh[threadIdx.x]  += sh[threadIdx.x + o];
      sh2[threadIdx.x] += sh2[threadIdx.x + o];
    }
    __syncthreads();
  }
  if (threadIdx.x == 0) {
    float mu  = sh[0] / (float)Mrows;
    float var = sh2[0] / (float)Mrows - mu * mu;
    mean[c] = mu;
    rstd[c] = rsqrtf(var + 1e-5f);
  }
}

// mode 0: out = relu(bn(t))
// mode 1: v = relu(bn(t) + hbuf); hbuf = v; ms += v
__global__ void bn_apply_k(const float* __restrict__ t,
                           const float* __restrict__ mean,
                           const float* __restrict__ rstd,
                           const float* __restrict__ g,
                           const float* __restrict__ bb,
                           float* __restrict__ out,
                           float* __restrict__ hbuf,
                           float* __restrict__ ms,
                           int total, int C, int mode) {
  int i = blockIdx.x * blockDim.x + threadIdx.x;
  if (i >= total) return;
  int c = i % C;
  float v = g[c] * (t[i] - mean[c]) * rstd[c] + bb[c];
  if (mode == 0) {
    v = v > 0.f ? v : 0.f;
    out[i] = v;
  } else {
    v += hbuf[i];
    v = v > 0.f ? v : 0.f;
    hbuf[i] = v;
    ms[i] += v;
  }
}

// ---------------------------------------------------------------------------
// 6) FC: emb[64,256] = relu(h_flat[64,8192] x fc_w^T + fc_b), NB=4
//    flat k = c*64 + y*8 + x over NCHW; h is NHWC [B,8,8,128]
// ---------------------------------------------------------------------------
__global__ void fc_k(const float* __restrict__ h, const float* __restrict__ w,
                     const float* __restrict__ bias, float* __restrict__ out) {
  const int Ngroups = 4;               // 256 / (16*4)
  int wid  = (blockIdx.x * blockDim.x + threadIdx.x) / WAVE;
  int lane = threadIdx.x & 31;
  if (wid >= 4 * Ngroups) return;      // M = 64 -> 4 tiles
  int tm = wid / Ngroups, tg = wid - tm * Ngroups;
  int row0 = tm << 4, col0 = tg << 6;  // 64 cols per wave
  int half = lane >> 4, n = lane & 15;
  int bA = row0 + n;                   // batch row of A

  v8f acc[4];
#pragma unroll
  for (int j = 0; j < 4; ++j) acc[j] = (v8f){};

  for (int k0 = 0; k0 < 8192; k0 += 4) {
    int k = k0 + 2 * half;
    int c0 = k >> 6,       rest0 = k & 63;
    int c1 = (k + 1) >> 6, rest1 = (k + 1) & 63;
    v2f av;
    av.x = h[((size_t)bA * 64 + rest0) * 128 + c0];
    av.y = h[((size_t)bA * 64 + rest1) * 128 + c1];
#pragma unroll
    for (int j = 0; j < 4; ++j) {
      const float* brow = w + (size_t)(col0 + j * 16 + n) * 8192;
      v2f bv = *(const v2f*)(brow + k);
      acc[j] = wmma4(av, bv, acc[j]);
    }
  }
#pragma unroll
  for (int j = 0; j < 4; ++j) {
    int oc = col0 + j * 16 + n;
#pragma unroll
    for (int i = 0; i < 8; ++i) {
      int m = row0 + i + 8 * half;
      float v = acc[j][i] + bias[oc];
      v = v > 0.f ? v : 0.f;
      out[(size_t)m * 256 + oc] = v;
    }
  }
}

// ---------------------------------------------------------------------------
// 7) map_skip NHWC [B,8,8,128] -> d_out NCHW [B,128,8,8]
// ---------------------------------------------------------------------------
__global__ void nhwc_to_nchw_k(const float* __restrict__ ms,
                               float* __restrict__ out) {
  int i = blockIdx.x * blockDim.x + threadIdx.x;
  if (i >= 64 * 128 * 64) return;
  int b = i >> 13;        // /8192
  int r = i & 8191;
  int c = r >> 6;         // /64
  int p = r & 63;         // y*8+x
  out[i] = ms[((size_t)b * 64 + p) * 128 + c];
}

// ---------------------------------------------------------------------------
extern "C" void kernel_launch(void* const* d_in, const int* in_sizes, int n_in,
                              void* d_out, int out_size, void* d_ws, size_t ws_size,
                              hipStream_t stream) {
  const float* x      = (const float*)d_in[0];   // [64,18,64,64]
  const float* ent    = (const float*)d_in[1];   // [64,512,256]
  const int*   xy     = (const int*)  d_in[2];   // [64,512,16]
  const float* c1w    = (const float*)d_in[3];   // [32,256]
  const float* pw     = (const float*)d_in[4];   // [32,50,1,1]
  const float* pb     = (const float*)d_in[5];
  const float* d1w    = (const float*)d_in[6];   // [64,32,4,4]
  const float* d1b    = (const float*)d_in[7];
  const float* d2w    = (const float*)d_in[8];   // [128,64,4,4]
  const float* d2b    = (const float*)d_in[9];
  const float* d3w    = (const float*)d_in[10];  // [128,128,4,4]
  const float* d3b    = (const float*)d_in[11];
  const float* rc1w   = (const float*)d_in[12];  // [4,128,128,3,3]
  const float* rb1g   = (const float*)d_in[13];
  const float* rb1b   = (const float*)d_in[14];
  const float* rc2w   = (const float*)d_in[15];
  const float* rb2g   = (const float*)d_in[16];
  const float* rb2b   = (const float*)d_in[17];
  const float* fcw    = (const float*)d_in[18];  // [256,8192]
  const float* fcb    = (const float*)d_in[19];
  float* out = (float*)d_out;

  // Workspace layout (floats). scat region (8,388,608 f) is reused by all
  // post-projection activations (exactly fills it).
  float* ws    = (float*)d_ws;
  float* scat  = ws;                         // [B,64,64,32]  8,388,608
  float* h2    = ws;                         // [B,32,32,64]  4,194,304 (aliases scat)
  float* h3    = ws + 4194304;               // [B,16,16,128] 2,097,152
  float* h4    = ws + 6291456;               // [B,8,8,128]     524,288
  float* t1    = ws + 6815744;               //                 524,288
  float* t2    = ws + 7340032;               //                 524,288
  float* msk   = ws + 7864320;               // map_skip        524,288
  float* red   = ws + 8388608;               // [32768,32]    1,048,576
  float* h1    = ws + 9437184;               // [B,64,64,32]  8,388,608
  float* mean  = ws + 17825792;              // 128
  float* rstd  = ws + 17825920;              // 128
  float* wpakA = ws + 17826048;              // entity weights    16,384
  float* wpakB = ws + 17842432;              // project weights    3,328
  float* wpak  = ws + 17845760;              // conv weights     524,288
                                             // end: 18,370,048 f (~73.5 MB)

  // 1) entity GEMM + relu  (repack conv1_w first: OC=32, IC=256, 1x1)
  repack_w_k<<<32, 256, 0, stream>>>(c1w, wpakA, 256, 32, 1, 1, 256);
  entity_gemm_k<<<256, 256, 0, stream>>>(ent, wpakA, red);

  // 2) zero scatter map, scatter-add
  hipMemsetAsync(scat, 0, (size_t)8388608 * sizeof(float), stream);
  scatter_k<<<64 * 512, 32, 0, stream>>>(xy, red, scat);

  // 3) projection 1x1 conv -> h1  (OC=32, K=50 -> Kpad=52)
  repack_w_k<<<7, 256, 0, stream>>>(pw, wpakB, 50, 32, 1, 1, 52);
  project_k<<<2048, 256, 0, stream>>>(scat, x, wpakB, pb, h1);

  // 4) downsample convs (scat region now free -> h2/h3/h4 alias it)
  repack_w_k<<<128, 256, 0, stream>>>(d1w, wpak, 32, 64, 4, 4, 512);
  conv_k<4><<<512, 256, 0, stream>>>(h1, 64, 64, 32, wpak, 64, d1b, h2, 32, 32,
                                     4, 4, 2, 1, 1, 65536);
  repack_w_k<<<512, 256, 0, stream>>>(d2w, wpak, 64, 128, 4, 4, 1024);
  conv_k<4><<<256, 256, 0, stream>>>(h2, 32, 32, 64, wpak, 128, d2b, h3, 16, 16,
                                     4, 4, 2, 1, 1, 16384);
  repack_w_k<<<1024, 256, 0, stream>>>(d3w, wpak, 128, 128, 4, 4, 2048);
  conv_k<4><<<64, 256, 0, stream>>>(h3, 16, 16, 128, wpak, 128, d3b, h4, 8, 8,
                                    4, 4, 2, 1, 1, 4096);

  // map_skip = h4
  hipMemcpyAsync(msk, h4, (size_t)524288 * sizeof(float),
                 hipMemcpyDeviceToDevice, stream);

  // 5) residual blocks
  for (int i = 0; i < 4; ++i) {
    const float* w1 = rc1w + (size_t)i * 128 * 128 * 9;
    const float* w2 = rc2w + (size_t)i * 128 * 128 * 9;
    const float* g1 = rb1g + (size_t)i * 128;  const float* b1 = rb1b + (size_t)i * 128;
    const float* g2 = rb2g + (size_t)i * 128;  const float* b2 = rb2b + (size_t)i * 128;

    repack_w_k<<<576, 256, 0, stream>>>(w1, wpak, 128, 128, 3, 3, 1152);
    conv_k<4><<<64, 256, 0, stream>>>(h4, 8, 8, 128, wpak, 128, nullptr, t1, 8, 8,
                                      3, 3, 1, 1, 0, 4096);
    bn_stats_k<<<128, 256, 0, stream>>>(t1, mean, rstd, 4096, 128);
    bn_apply_k<<<2048, 256, 0, stream>>>(t1, mean, rstd, g1, b1, t2, nullptr,
                                         nullptr, 524288, 128, 0);
    repack_w_k<<<576, 256, 0, stream>>>(w2, wpak, 128, 128, 3, 3, 1152);
    conv_k<4><<<64, 256, 0, stream>>>(t2, 8, 8, 128, wpak, 128, nullptr, t1, 8, 8,
                                      3, 3, 1, 1, 0, 4096);
    bn_stats_k<<<128, 256, 0, stream>>>(t1, mean, rstd, 4096, 128);
    bn_apply_k<<<2048, 256, 0, stream>>>(t1, mean, rstd, g2, b2, nullptr, h4,
                                         msk, 524288, 128, 1);
  }

  // 6) outputs: map_skip (NCHW) then embedded_spatial
  nhwc_to_nchw_k<<<2048, 256, 0, stream>>>(msk, out);
  fc_k<<<2, 256, 0, stream>>>(h4, fcw, fcb, out + 524288);
}